// EglGAT_56599079026979
// MI455X (gfx1250) — compile-verified
//
#include <hip/hip_runtime.h>
#include <math.h>

#define NN 50000
#define EE 800000

typedef __attribute__((ext_vector_type(2))) float v2f;
typedef __attribute__((ext_vector_type(8))) float v8f;

// ---------------- order-preserving float <-> uint for atomicMax ----------------
__device__ __forceinline__ unsigned f2ord(float f) {
  unsigned u = __float_as_uint(f);
  return (u & 0x80000000u) ? ~u : (u | 0x80000000u);
}
__device__ __forceinline__ float ord2f(unsigned e) {
  unsigned u = (e & 0x80000000u) ? (e ^ 0x80000000u) : ~e;
  return __uint_as_float(u);
}

// ---------------- GEMM: OUT[NN x Oout] = X[NN x K] @ W[Opad x K]^T -------------
// WMMA f32 16x16x4, one wave per 16-row x 64-col tile.
// Software-pipelined K loop: register double-buffered fragments, so loads for
// step k+4 are in flight while WMMAs consume step k (no per-WMMA loadcnt-0 wait).
// A 16x4 f32 layout: lane(0..15)->M=lane,K={0,1}; lane(16..31)->M=lane-16,K={2,3}
// B 4x16  f32 layout: lane(0..15)->N=lane,K={0,1}; lane(16..31)->N=lane-16,K={2,3}
// C/D 16x16 f32: col = lane&15, row = vgpr + 8*(lane>=16)
// Requires: Opad % 64 == 0 (weights zero-padded), K % 8 == 0, K >= 8, NN % 16 == 0.
__global__ __launch_bounds__(32)
void gat_gemm_xwt(const float* __restrict__ X, const float* __restrict__ W,
                  float* __restrict__ OUT, int K, int Opad, int Oout) {
  const int lane = threadIdx.x & 31;
  const int row0 = blockIdx.x << 4;
  const int col0 = blockIdx.y << 6;
  const int mr   = lane & 15;
  const int half = lane >> 4;
  const int koff = half << 1;

  const float* xrow = X + (size_t)(row0 + mr) * (size_t)K + koff;
  const float* wrow = W + (size_t)(col0 + mr) * (size_t)K + koff;
  const size_t wstep = (size_t)16 * (size_t)K;   // 16 output cols per B tile

  v8f acc[4] = {};
  v2f a0, a1;
  v2f b0[4], b1[4];

  // prologue: fragments for k = 0
  a0 = *(const v2f*)(xrow);
#pragma unroll
  for (int t = 0; t < 4; ++t) b0[t] = *(const v2f*)(wrow + (size_t)t * wstep);

  int k0 = 0;
  for (; k0 + 8 < K; k0 += 8) {
    // issue loads for k0+4 while consuming k0
    a1 = *(const v2f*)(xrow + k0 + 4);
#pragma unroll
    for (int t = 0; t < 4; ++t)
      b1[t] = *(const v2f*)(wrow + (size_t)t * wstep + k0 + 4);
#pragma unroll
    for (int t = 0; t < 4; ++t)
      acc[t] = __builtin_amdgcn_wmma_f32_16x16x4_f32(
          false, a0, false, b0[t], (short)0, acc[t], false, false);

    // issue loads for k0+8 while consuming k0+4
    a0 = *(const v2f*)(xrow + k0 + 8);
#pragma unroll
    for (int t = 0; t < 4; ++t)
      b0[t] = *(const v2f*)(wrow + (size_t)t * wstep + k0 + 8);
#pragma unroll
    for (int t = 0; t < 4; ++t)
      acc[t] = __builtin_amdgcn_wmma_f32_16x16x4_f32(
          false, a1, false, b1[t], (short)0, acc[t], false, false);
  }

  // epilogue: k0 == K-8; b0 holds step K-8, load step K-4, consume both
  a1 = *(const v2f*)(xrow + k0 + 4);
#pragma unroll
  for (int t = 0; t < 4; ++t)
    b1[t] = *(const v2f*)(wrow + (size_t)t * wstep + k0 + 4);
#pragma unroll
  for (int t = 0; t < 4; ++t)
    acc[t] = __builtin_amdgcn_wmma_f32_16x16x4_f32(
        false, a0, false, b0[t], (short)0, acc[t], false, false);
#pragma unroll
  for (int t = 0; t < 4; ++t)
    acc[t] = __builtin_amdgcn_wmma_f32_16x16x4_f32(
        false, a1, false, b1[t], (short)0, acc[t], false, false);

#pragma unroll
  for (int t = 0; t < 4; ++t) {
    const int oc = col0 + (t << 4) + mr;
    if (oc < Oout) {
      float* orow = OUT + (size_t)(row0 + (half << 3)) * (size_t)Oout + oc;
#pragma unroll
      for (int v = 0; v < 8; ++v) {
        orow[(size_t)v * (size_t)Oout] = acc[t][v];
      }
    }
  }
}

// ---------------- zero-pad weight rows: Wdst[padrows x K] <- Wsrc[rows x K] ----
__global__ void gat_pad_w(const float* __restrict__ Wsrc, float* __restrict__ Wdst,
                          int rows, int padrows, int K) {
  int idx = blockIdx.x * blockDim.x + threadIdx.x;
  int total = padrows * K;
  if (idx >= total) return;
  int r = idx / K;
  Wdst[idx] = (r < rows) ? Wsrc[idx] : 0.0f;
}

// ---------------- el/er: per (node, head) dot of H row-slice with al/ar -------
__global__ void gat_el_er(const float* __restrict__ Hm, const float* __restrict__ al,
                          const float* __restrict__ ar, float* __restrict__ el,
                          float* __restrict__ er, int total, int Hh, int F) {
  int idx = blockIdx.x * blockDim.x + threadIdx.x;
  if (idx >= total) return;
  int h = idx % Hh;
  int n = idx / Hh;
  const float* row = Hm + (size_t)n * (size_t)(Hh * F) + (size_t)h * F;
  const float* a = al + h * F;
  const float* b = ar + h * F;
  float sl = 0.f, sr = 0.f;
  for (int f = 0; f < F; ++f) {
    float v = row[f];
    sl += v * a[f];
    sr += v * b[f];
  }
  el[idx] = sl;
  er[idx] = sr;
}

// ---------------- init segment max (encoded) / denom --------------------------
__global__ void gat_init_nm(unsigned* __restrict__ mEnc, float* __restrict__ den,
                            int total) {
  int idx = blockIdx.x * blockDim.x + threadIdx.x;
  if (idx < total) { mEnc[idx] = 0u; den[idx] = 0.0f; }
}

// ---------------- edge score + atomic segment-max -----------------------------
__global__ void gat_edge_smax(const float* __restrict__ el, const float* __restrict__ er,
                              const int* __restrict__ src, const int* __restrict__ dst,
                              float* __restrict__ sb, unsigned* __restrict__ mEnc,
                              int total, int Hh) {
  int idx = blockIdx.x * blockDim.x + threadIdx.x;
  if (idx >= total) return;
  int e = idx / Hh;
  int h = idx - e * Hh;
  int s_ = src[e], d_ = dst[e];
  float s = el[s_ * Hh + h] + er[d_ * Hh + h];
  s = (s > 0.0f) ? s : 0.2f * s;             // leaky_relu, NEG_SLOPE=0.2
  sb[idx] = s;
  atomicMax(mEnc + d_ * Hh + h, f2ord(s));
}

// ---------------- exp(s - m) + atomic segment-sum -----------------------------
__global__ void gat_edge_expsum(float* __restrict__ sb, const unsigned* __restrict__ mEnc,
                                float* __restrict__ den, const int* __restrict__ dst,
                                int total, int Hh) {
  int idx = blockIdx.x * blockDim.x + threadIdx.x;
  if (idx >= total) return;
  int e = idx / Hh;
  int h = idx - e * Hh;
  int d_ = dst[e];
  float m = ord2f(mEnc[d_ * Hh + h]);
  float ex = expf(sb[idx] - m);
  sb[idx] = ex;
  atomicAdd(den + d_ * Hh + h, ex);
}

// ---------------- alpha-weighted scatter: AGG[dst] += H[src] * alpha ----------
__global__ void gat_scatter(const float* __restrict__ Hm, const float* __restrict__ sb,
                            const float* __restrict__ den, const int* __restrict__ src,
                            const int* __restrict__ dst, float* __restrict__ AGG,
                            int HF, int F, int Hh) {
  int e = blockIdx.x;
  int j = threadIdx.x;
  if (j >= HF) return;
  int s_ = src[e], d_ = dst[e];
  int h = j / F;
  float alpha = sb[(size_t)e * Hh + h] / den[(size_t)d_ * Hh + h];
  atomicAdd(AGG + (size_t)d_ * HF + j, Hm[(size_t)s_ * HF + j] * alpha);
}

// ---------------- elementwise ELU (in place) ----------------------------------
__global__ void gat_elu(float* __restrict__ y, long total) {
  long idx = (long)blockIdx.x * blockDim.x + threadIdx.x;
  if (idx >= total) return;
  float x = y[idx];
  y[idx] = (x > 0.0f) ? x : expm1f(x);
}

// ---------------- one GAT layer -----------------------------------------------
static void run_layer(const float* X, int K, int Hh, int F,
                      const float* W, const float* al, const float* ar, const float* rW,
                      const int* src, const int* dst,
                      float* Hbuf, float* AGG, bool act, int Opad,
                      float* el, float* er, float* den, unsigned* mEnc, float* sb,
                      hipStream_t stream) {
  const int HF = Hh * F;
  dim3 ggrid(NN / 16, Opad / 64);
  gat_gemm_xwt<<<ggrid, 32, 0, stream>>>(X, W, Hbuf, K, Opad, HF);  // h   = x @ W^T
  gat_gemm_xwt<<<ggrid, 32, 0, stream>>>(X, rW, AGG, K, Opad, HF);  // agg = x @ rW^T

  const int nh = NN * Hh;
  gat_el_er<<<(nh + 255) / 256, 256, 0, stream>>>(Hbuf, al, ar, el, er, nh, Hh, F);
  gat_init_nm<<<(nh + 255) / 256, 256, 0, stream>>>(mEnc, den, nh);

  const int eh = EE * Hh;
  gat_edge_smax<<<(eh + 255) / 256, 256, 0, stream>>>(el, er, src, dst, sb, mEnc, eh, Hh);
  gat_edge_expsum<<<(eh + 255) / 256, 256, 0, stream>>>(sb, mEnc, den, dst, eh, Hh);

  const int bs = (HF >= 256) ? 256 : 64;
  gat_scatter<<<EE, bs, 0, stream>>>(Hbuf, sb, den, src, dst, AGG, HF, F, Hh);

  if (act) {
    long tot = (long)NN * HF;
    gat_elu<<<(unsigned)((tot + 255) / 256), 256, 0, stream>>>(AGG, tot);
  }
}

extern "C" void kernel_launch(void* const* d_in, const int* in_sizes, int n_in,
                              void* d_out, int out_size, void* d_ws, size_t ws_size,
                              hipStream_t stream) {
  (void)in_sizes; (void)n_in; (void)out_size; (void)ws_size;

  const float* feat = (const float*)d_in[0];
  const int*   src  = (const int*)d_in[1];
  const int*   dst  = (const int*)d_in[2];
  const float* W0  = (const float*)d_in[3];
  const float* al0 = (const float*)d_in[4];
  const float* ar0 = (const float*)d_in[5];
  const float* rW0 = (const float*)d_in[6];
  const float* W1  = (const float*)d_in[7];
  const float* al1 = (const float*)d_in[8];
  const float* ar1 = (const float*)d_in[9];
  const float* rW1 = (const float*)d_in[10];
  const float* W2  = (const float*)d_in[11];
  const float* al2 = (const float*)d_in[12];
  const float* ar2 = (const float*)d_in[13];
  const float* rW2 = (const float*)d_in[14];

  // workspace layout (floats)
  const size_t NF = (size_t)NN * 256;
  float* ws    = (float*)d_ws;
  float* bufH  = ws;                         // [NN,256] h of current layer
  float* bufY0 = bufH  + NF;                 // [NN,256] layer-0 output
  float* bufY1 = bufY0 + NF;                 // [NN,256] layer-1 output
  float* el    = bufY1 + NF;                 // [NN,4]
  float* er    = el  + (size_t)NN * 4;       // [NN,4]
  float* den   = er  + (size_t)NN * 4;       // [NN,4]
  unsigned* mE = (unsigned*)(den + (size_t)NN * 4);  // [NN,4]
  float* sb    = (float*)mE + (size_t)NN * 4;        // [EE,4]
  float* W2p   = sb + (size_t)EE * 4;        // [64,256] zero-padded W2
  float* rW2p  = W2p + (size_t)64 * 256;     // [64,256] zero-padded rW2

  // zero-pad layer-2 weights (40x256 -> 64x256) so the GEMM K-loop is branch-free
  {
    int tot = 64 * 256;
    gat_pad_w<<<(tot + 255) / 256, 256, 0, stream>>>(W2,  W2p,  40, 64, 256);
    gat_pad_w<<<(tot + 255) / 256, 256, 0, stream>>>(rW2, rW2p, 40, 64, 256);
  }

  // layer 0: in=128, heads=4, F=64, ELU
  run_layer(feat, 128, 4, 64, W0, al0, ar0, rW0, src, dst,
            bufH, bufY0, true, 256, el, er, den, mE, sb, stream);
  // layer 1: in=256, heads=4, F=64, ELU
  run_layer(bufY0, 256, 4, 64, W1, al1, ar1, rW1, src, dst,
            bufH, bufY1, true, 256, el, er, den, mE, sb, stream);
  // layer 2: in=256, heads=1, F=40 (padded to 64), no activation
  run_layer(bufY1, 256, 1, 40, W2p, al2, ar2, rW2p, src, dst,
            bufH, (float*)d_out, false, 64, el, er, den, mE, sb, stream);
}